// Interp2MaskBinary_74929999446522
// MI455X (gfx1250) — compile-verified
//
#include <hip/hip_runtime.h>
#include <cstdint>

// Problem dims (fixed by the reference)
constexpr int Bn  = 8, Cn = 64, Hn = 256, Wn = 256;
constexpr int HW  = Hn * Wn;        // 65536
constexpr int CHW = Cn * HW;        // 4194304
constexpr int Q   = Bn * HW;        // 524288

constexpr int BQ    = 256;          // queries per tile == W (one image row), 8 wave32/block
constexpr int TILES = 2;            // double-buffered coordinate staging

// ---- CDNA5 async global->LDS copy (ASYNCcnt tracked, §15.18 op 96) ----
__device__ __forceinline__ void async_f32_to_lds(const float* g, float* lds) {
    // low 32 bits of a generic pointer into the LDS aperture == DS byte address
    uint32_t lds_addr = (uint32_t)(uintptr_t)lds;
    uint64_t gaddr    = (uint64_t)(uintptr_t)g;
    asm volatile("global_load_async_to_lds_b32 %0, %1, off"
                 :: "v"(lds_addr), "v"(gaddr) : "memory");
}
#define WAIT_ASYNCCNT_2() asm volatile("s_wait_asynccnt 2" ::: "memory")
#define WAIT_ASYNCCNT_0() asm volatile("s_wait_asynccnt 0" ::: "memory")

__device__ __forceinline__ int iclamp(int v, int lo, int hi) {
    return v < lo ? lo : (v > hi ? hi : v);
}

__global__ __launch_bounds__(BQ)
void interp2_mask_kernel(const float* __restrict__ v,
                         const float* __restrict__ xq,
                         const float* __restrict__ yq,
                         const float* __restrict__ mask,
                         float* __restrict__ out,
                         float* __restrict__ valid)
{
    __shared__ float s_x[TILES][BQ];
    __shared__ float s_y[TILES][BQ];

    const int tid  = threadIdx.x;
    const int base = blockIdx.x * (BQ * TILES);

    // Stage both tiles' coordinates asynchronously (4 in-order async ops).
    async_f32_to_lds(&xq[base + tid],      &s_x[0][tid]);
    async_f32_to_lds(&yq[base + tid],      &s_y[0][tid]);
    async_f32_to_lds(&xq[base + BQ + tid], &s_x[1][tid]);
    async_f32_to_lds(&yq[base + BQ + tid], &s_y[1][tid]);

    #pragma unroll
    for (int t = 0; t < TILES; ++t) {
        // Async loads complete in order: <=2 outstanding means tile0's 2 are done.
        if (t == 0) { WAIT_ASYNCCNT_2(); } else { WAIT_ASYNCCNT_0(); }

        const int   q  = base + t * BQ + tid;
        const float x  = s_x[t][tid];
        const float y  = s_y[t][tid];
        const int   b  = q >> 16;          // / HW
        const int   hw = q & (HW - 1);     // h*W + w

        // Corner coords: floor, then clamp (matches reference order).
        const int x0 = iclamp((int)floorf(x), 0, Wn - 1);
        const int y0 = iclamp((int)floorf(y), 0, Hn - 1);
        const int x1 = x0 + 1 > Wn - 1 ? Wn - 1 : x0 + 1;
        const int y1 = y0 + 1 > Hn - 1 ? Hn - 1 : y0 + 1;

        const float fx  = x - (float)x0;
        const float fy  = y - (float)y0;
        const float w00 = (1.0f - fy) * (1.0f - fx);
        const float w01 = (1.0f - fy) * fx;
        const float w10 = fy * (1.0f - fx);
        const float w11 = fy * fx;

        const int o00 = y0 * Wn + x0, o01 = y0 * Wn + x1;
        const int o10 = y1 * Wn + x0, o11 = y1 * Wn + x1;

        // Mask gathers (256 KB/image -> L2/WGP$ resident).
        const float* mb  = mask + (size_t)b * HW;
        const float  m00 = mb[o00], m01 = mb[o01], m10 = mb[o10], m11 = mb[o11];

        const float mw       = m00 * w00 + m01 * w01 + m10 * w10 + m11 * w11;
        const float inv_den  = 1.0f / (mw + 1e-12f);
        const bool  inv_mask = ((1.0f - mw) * inv_den) > 0.5f;
        const bool  oob      = (x < 0.0f) | (x >= (float)Wn) |
                               (y < 0.0f) | (y >= (float)Hn);
        const float scale    = (oob || inv_mask) ? 0.0f : inv_den;

        // Fold mask * weight * normalization * invalidation into 4 coefficients.
        const float a00 = m00 * w00 * scale;
        const float a01 = m01 * w01 * scale;
        const float a10 = m10 * w10 * scale;
        const float a11 = m11 * w11 * scale;

        // valid = 1 - inv_mask (reference excludes the OOB term here). NT store:
        // output is write-once, keep L2 for v.
        __builtin_nontemporal_store(inv_mask ? 0.0f : 1.0f, &valid[q]);

        const float* vb = v   + (size_t)b * CHW;
        float*       ob = out + (size_t)b * CHW + hw;

        // 64-channel gather+blend. Unroll x4 -> 16 L2 gathers in flight/wave.
        #pragma unroll 4
        for (int c = 0; c < Cn; ++c) {
            const float* vc = vb + (size_t)c * HW;
            float r = a00 * vc[o00];
            r = fmaf(a01, vc[o01], r);
            r = fmaf(a10, vc[o10], r);
            r = fmaf(a11, vc[o11], r);
            __builtin_nontemporal_store(r, ob + (size_t)c * HW);
        }
    }
}

extern "C" void kernel_launch(void* const* d_in, const int* in_sizes, int n_in,
                              void* d_out, int out_size, void* d_ws, size_t ws_size,
                              hipStream_t stream) {
    const float* v    = (const float*)d_in[0];
    const float* xq   = (const float*)d_in[1];
    const float* yq   = (const float*)d_in[2];
    const float* mask = (const float*)d_in[3];

    float* out   = (float*)d_out;
    float* valid = out + (size_t)Bn * Cn * Hn * Wn;   // tuple order: (out, valid)

    const int blocks = Q / (BQ * TILES);              // 1024
    hipLaunchKernelGGL(interp2_mask_kernel, dim3(blocks), dim3(BQ), 0, stream,
                       v, xq, yq, mask, out, valid);
}